// Rcell_29094108463784
// MI455X (gfx1250) — compile-verified
//
#include <hip/hip_runtime.h>

#define T_STEPS 8192
#define BATCH   1024
#define DT      1e-3f

typedef __attribute__((ext_vector_type(2))) float v2f;
typedef __attribute__((ext_vector_type(8))) float v8f;

// ---- workspace layout (float offsets) ----
#define OFF_W    0                       // 256: 16x16 Toeplitz conv matrix, row-major
#define OFF_HA   256                     // 32 : Haug 16x2 (DT*mB*M^t)
#define OFF_M    288                     // 4  : M
#define OFF_M8   292                     // 4  : M^8
#define OFF_M512 296                     // 4  : M^512
#define OFF_XI   300                     // 4  : XiCov
#define OFF_V8   512                     // 1024*1024*2 : 8-step chunk aggregates
#define OFF_V512 (OFF_V8 + 2097152)      // 16*1024*2   : 512-step aggregates
#define OFF_SJ   (OFF_V512 + 32768)      // 16*1024*2   : superchunk start states
#define OFF_S8   (OFF_SJ + 32768)        // 1024*1024*2 : 8-step-block start states

__device__ inline void mm2(float* C, const float* A, const float* B) {
  float c0 = A[0]*B[0] + A[1]*B[2];
  float c1 = A[0]*B[1] + A[1]*B[3];
  float c2 = A[2]*B[0] + A[3]*B[2];
  float c3 = A[2]*B[1] + A[3]*B[3];
  C[0]=c0; C[1]=c1; C[2]=c2; C[3]=c3;
}

// K0: build constants (single thread; trivial cost)
__global__ void k_setup(const float* __restrict__ Ain,
                        const float* __restrict__ Bin,
                        const float* __restrict__ Xin,
                        float* __restrict__ ws) {
  if (threadIdx.x != 0 || blockIdx.x != 0) return;
  const float s2 = 1.41421356237309504880f;
  float Xi[4] = {Xin[0], Xin[1], Xin[2], Xin[3]};
  float mB[4];                               // -sqrt(2) * B^T
  mB[0] = -s2*Bin[0]; mB[1] = -s2*Bin[2];
  mB[2] = -s2*Bin[1]; mB[3] = -s2*Bin[3];
  float XC[4]; mm2(XC, Xi, mB);              // XiCov @ mB
  float M[4];                                // I + DT*(A - XiCov@mB)
  M[0] = 1.f + DT*(Ain[0]-XC[0]); M[1] = DT*(Ain[1]-XC[1]);
  M[2] = DT*(Ain[2]-XC[2]);       M[3] = 1.f + DT*(Ain[3]-XC[3]);
  float DTmB[4] = {DT*mB[0], DT*mB[1], DT*mB[2], DT*mB[3]};
  float P[9][4];
  P[0][0]=1.f; P[0][1]=0.f; P[0][2]=0.f; P[0][3]=1.f;
  for (int k=1;k<=8;k++) mm2(P[k], M, P[k-1]);
  float M512[4] = {P[8][0],P[8][1],P[8][2],P[8][3]};
  for (int q=0;q<6;q++) mm2(M512, M512, M512);           // (M^8)^(2^6) = M^512
  for (int r=0;r<256;r++) ws[OFF_W + r] = 0.f;
  for (int t=0;t<8;t++) {
    for (int k=0;k<t;k++) {
      float t1[4], t2[4];
      mm2(t1, DTmB, P[t-1-k]);
      mm2(t2, t1, Xi);                                   // DT*mB*M^{t-1-k}*XiCov
      ws[OFF_W + (2*t+0)*16 + 2*k+0] = t2[0];
      ws[OFF_W + (2*t+0)*16 + 2*k+1] = t2[1];
      ws[OFF_W + (2*t+1)*16 + 2*k+0] = t2[2];
      ws[OFF_W + (2*t+1)*16 + 2*k+1] = t2[3];
    }
    float h[4]; mm2(h, DTmB, P[t]);                      // DT*mB*M^t
    ws[OFF_HA + (2*t+0)*2 + 0] = h[0];
    ws[OFF_HA + (2*t+0)*2 + 1] = h[1];
    ws[OFF_HA + (2*t+1)*2 + 0] = h[2];
    ws[OFF_HA + (2*t+1)*2 + 1] = h[3];
  }
  for (int i=0;i<4;i++) {
    ws[OFF_M+i]    = M[i];
    ws[OFF_M8+i]   = P[8][i];
    ws[OFF_M512+i] = M512[i];
    ws[OFF_XI+i]   = Xi[i];
  }
}

// K1: 8-step chunk aggregates v8[j][b] = sum_i M^{7-i} * Xi*dy[8j+i][b]  (1M threads)
__global__ __launch_bounds__(256) void k_v8(const float2* __restrict__ dy2,
                                            float* __restrict__ ws) {
  int tid = blockIdx.x*256 + threadIdx.x;
  int j = tid >> 10, b = tid & 1023;
  float M00=ws[OFF_M], M01=ws[OFF_M+1], M10=ws[OFF_M+2], M11=ws[OFF_M+3];
  float X00=ws[OFF_XI],X01=ws[OFF_XI+1],X10=ws[OFF_XI+2],X11=ws[OFF_XI+3];
  float v0=0.f, v1=0.f;
  int base = (j<<13) + b;
  #pragma unroll
  for (int i=0;i<8;i++) {
    float2 u = dy2[base + (i<<10)];
    float u0 = X00*u.x + X01*u.y;
    float u1 = X10*u.x + X11*u.y;
    float n0 = M00*v0 + M01*v1 + u0;
    float n1 = M10*v0 + M11*v1 + u1;
    v0=n0; v1=n1;
  }
  ((float2*)(ws + OFF_V8))[(j<<10) + b] = make_float2(v0, v1);
}

// K2: combine 64 v8's into 512-step aggregates (16384 threads)
__global__ __launch_bounds__(256) void k_v512(float* __restrict__ ws) {
  int tid = blockIdx.x*256 + threadIdx.x;
  int J = tid >> 10, b = tid & 1023;
  float M00=ws[OFF_M8],M01=ws[OFF_M8+1],M10=ws[OFF_M8+2],M11=ws[OFF_M8+3];
  const float2* v8 = (const float2*)(ws + OFF_V8);
  float w0=0.f, w1=0.f;
  for (int q=0;q<64;q++) {
    float2 v = v8[((J*64+q)<<10) + b];
    float n0 = M00*w0 + M01*w1 + v.x;
    float n1 = M10*w0 + M11*w1 + v.y;
    w0=n0; w1=n1;
  }
  ((float2*)(ws+OFF_V512))[(J<<10)+b] = make_float2(w0,w1);
}

// K3: scan 16 superchunks per trajectory -> superchunk start states (1024 threads)
__global__ __launch_bounds__(256) void k_scanJ(const float2* __restrict__ sts0,
                                               float* __restrict__ ws) {
  int b = blockIdx.x*256 + threadIdx.x;
  float M00=ws[OFF_M512],M01=ws[OFF_M512+1],M10=ws[OFF_M512+2],M11=ws[OFF_M512+3];
  const float2* V = (const float2*)(ws+OFF_V512);
  float2* SJ = (float2*)(ws+OFF_SJ);
  float2 s = sts0[b];
  float s0c=s.x, s1c=s.y;
  for (int J=0;J<16;J++) {
    SJ[(J<<10)+b] = make_float2(s0c,s1c);
    float2 v = V[(J<<10)+b];
    float n0 = M00*s0c + M01*s1c + v.x;
    float n1 = M10*s0c + M11*s1c + v.y;
    s0c=n0; s1c=n1;
  }
}

// K4: redistribute -> start state of every 8-step block (16384 threads)
__global__ __launch_bounds__(256) void k_bound8(float* __restrict__ ws) {
  int tid = blockIdx.x*256 + threadIdx.x;
  int J = tid >> 10, b = tid & 1023;
  float M00=ws[OFF_M8],M01=ws[OFF_M8+1],M10=ws[OFF_M8+2],M11=ws[OFF_M8+3];
  const float2* v8 = (const float2*)(ws + OFF_V8);
  const float2* SJ = (const float2*)(ws + OFF_SJ);
  float2* s8 = (float2*)(ws + OFF_S8);
  float2 s = SJ[(J<<10)+b];
  float s0c=s.x, s1c=s.y;
  for (int q=0;q<64;q++) {
    int j = J*64 + q;
    s8[(j<<10)+b] = make_float2(s0c,s1c);
    float2 v = v8[(j<<10)+b];
    float n0 = M00*s0c + M01*s1c + v.x;
    float n1 = M10*s0c + M11*s1c + v.y;
    s0c=n0; s1c=n1;
  }
}

// K5: outputs via WMMA. One wave = (8 timesteps x 16 trajectories) tile:
//   Y[16x16] = W[16x16] @ U[16x16] + HaugPad[16x4] @ S0pad[4x16]
//   (4 K=4 slices for W@U, 1 padded slice for the boundary-state term)
__global__ __launch_bounds__(256) void k_outs(const float2* __restrict__ dy2,
                                              const float* __restrict__ ws,
                                              float* __restrict__ out) {
  int wave = blockIdx.x*8 + (threadIdx.x>>5);
  int lane = threadIdx.x & 31;
  int j  = wave >> 6;            // time block 0..1023 (8 steps each)
  int b0 = (wave & 63) << 4;     // 16-trajectory column base
  int t0 = j << 3;
  int half = lane >> 4;          // lanes 16-31 hold K+2 / rows M+8 per ISA layout
  int m    = lane & 15;

  const float* W  = ws + OFF_W;
  const float* HA = ws + OFF_HA;

  // A operands: lane m holds row M=m; VGPR0 = K{0|2}, VGPR1 = K{1|3} of each slice
  v2f a0,a1,a2,a3,ah;
  int wrow = m*16 + 2*half;
  a0.x = W[wrow+0];  a0.y = W[wrow+1];
  a1.x = W[wrow+4];  a1.y = W[wrow+5];
  a2.x = W[wrow+8];  a2.y = W[wrow+9];
  a3.x = W[wrow+12]; a3.y = W[wrow+13];
  ah.x = half ? 0.f : HA[m*2+0];       // cols 2..3 of the padded slice are zero
  ah.y = half ? 0.f : HA[m*2+1];

  // B operands: rows = (step,comp) pairs of raw dy; lanes 0-15 = low K half
  v2f b0v,b1v,b2v,b3v,bsv;
  float2 d0 = dy2[((t0+0+half)<<10) + b0 + m];
  float2 d1 = dy2[((t0+2+half)<<10) + b0 + m];
  float2 d2 = dy2[((t0+4+half)<<10) + b0 + m];
  float2 d3 = dy2[((t0+6+half)<<10) + b0 + m];
  b0v.x=d0.x; b0v.y=d0.y;
  b1v.x=d1.x; b1v.y=d1.y;
  b2v.x=d2.x; b2v.y=d2.y;
  b3v.x=d3.x; b3v.y=d3.y;
  const float2* s8 = (const float2*)(ws + OFF_S8);
  float2 s0 = s8[(j<<10) + b0 + m];
  bsv.x = half ? 0.f : s0.x;
  bsv.y = half ? 0.f : s0.y;

  v8f acc = {};
  acc = __builtin_amdgcn_wmma_f32_16x16x4_f32(false, a0, false, b0v, (short)0, acc, false, false);
  acc = __builtin_amdgcn_wmma_f32_16x16x4_f32(false, a1, false, b1v, (short)0, acc, false, false);
  acc = __builtin_amdgcn_wmma_f32_16x16x4_f32(false, a2, false, b2v, (short)0, acc, false, false);
  acc = __builtin_amdgcn_wmma_f32_16x16x4_f32(false, a3, false, b3v, (short)0, acc, false, false);
  acc = __builtin_amdgcn_wmma_f32_16x16x4_f32(false, ah, false, bsv, (short)0, acc, false, false);

  // D layout: VGPR v -> rows v (lanes 0-15) / v+8 (lanes 16-31), col = lane%16
  #pragma unroll
  for (int v=0; v<8; v++) {
    int r  = v + 8*half;
    int t  = r >> 1;
    int ic = r & 1;
    out[((((t0+t)<<10) + b0 + m)<<1) + ic] = acc[v];
  }
}

extern "C" void kernel_launch(void* const* d_in, const int* in_sizes, int n_in,
                              void* d_out, int out_size, void* d_ws, size_t ws_size,
                              hipStream_t stream) {
  // inputs: 0=dy, 1=sts0, 2=cov0(dead), 3=A, 4=D(dead), 5=E(dead), 6=B, 7=XiCov
  const float2* dy2  = (const float2*)d_in[0];
  const float2* sts0 = (const float2*)d_in[1];
  const float*  A    = (const float*)d_in[3];
  const float*  Bm   = (const float*)d_in[6];
  const float*  Xi   = (const float*)d_in[7];
  float* ws  = (float*)d_ws;
  float* out = (float*)d_out;

  k_setup <<<1,    1,   0, stream>>>(A, Bm, Xi, ws);
  k_v8    <<<4096, 256, 0, stream>>>(dy2, ws);
  k_v512  <<<64,   256, 0, stream>>>(ws);
  k_scanJ <<<4,    256, 0, stream>>>(sts0, ws);
  k_bound8<<<64,   256, 0, stream>>>(ws);
  k_outs  <<<8192, 256, 0, stream>>>(dy2, ws, out);

  (void)in_sizes; (void)n_in; (void)out_size; (void)ws_size;
}